// CornerNet_Loss_90692529422980
// MI455X (gfx1250) — compile-verified
//
#include <hip/hip_runtime.h>
#include <cstdint>

// CornerNet loss for MI455X (gfx1250, wave32).
// Roofline: 335MB streamed / 23.3 TB/s ~= 14.4us floor -> memory bound.
// Strategy: async global->LDS double-buffered streaming (CDNA5 TDM-lite path,
// ASYNCcnt), f32 WMMA (16x16x4) for cross-lane reductions, tiny epilogue kernels.

#define EPSV 1e-4f
#define BATCH 16
#define CCH 80
#define HH 128
#define WW 128
#define KK 128
#define HWSZ (HH * WW)
#define NPB (CCH * HH * WW)        // elements per batch per heatmap = 1310720
#define N4PB (NPB / 4)             // float4 per batch = 327680
#define BLKX 80                    // blocks per (batch, stream)
#define TPB 256
#define STRIDE4 (BLKX * TPB)       // 20480 float4 lanes per batch
#define ITERS (N4PB / STRIDE4)     // 16

typedef float v2f __attribute__((ext_vector_type(2)));
typedef float v8f __attribute__((ext_vector_type(8)));

// Full 32-lane f32 sum. Stage 1: V_WMMA_F32_16X16X4_F32 with A=ones folds
// lanes L and L+16 (B vgpr0 rows K0/K1 hold the lane values, D = colsum).
// Stage 2: 4 xor-shuffles across the remaining 16. EXEC must be all ones.
__device__ __forceinline__ float wave_sum32(float v) {
#if __has_builtin(__builtin_amdgcn_wmma_f32_16x16x4_f32)
  v2f a; a[0] = 1.0f; a[1] = 1.0f;
  v2f b; b[0] = v;    b[1] = 0.0f;
  v8f c = {};
  c = __builtin_amdgcn_wmma_f32_16x16x4_f32(false, a, false, b, (short)0, c,
                                            false, false);
  float s = c[0];   // lane L holds v[L%16] + v[L%16+16]
#else
  float s = v + __shfl_xor(v, 16, 32);
#endif
  s += __shfl_xor(s, 8, 32);
  s += __shfl_xor(s, 4, 32);
  s += __shfl_xor(s, 2, 32);
  s += __shfl_xor(s, 1, 32);
  return s;
}

// CDNA5 async global->LDS copy of 16B per lane; tracked by ASYNCcnt.
__device__ __forceinline__ void async_b128(uint32_t lds_addr, const void* gaddr) {
  asm volatile("global_load_async_to_lds_b128 %0, %1, off"
               :: "v"(lds_addr), "v"(gaddr) : "memory");
}
__device__ __forceinline__ void wait_async_le2() {
  asm volatile("s_wait_asynccnt 0x2" ::: "memory");
}
__device__ __forceinline__ void wait_async_0() {
  asm volatile("s_wait_asynccnt 0x0" ::: "memory");
}

// Stable log-sigmoid focal terms.
__device__ __forceinline__ void focal_elem(float t, float x, float& pos,
                                           float& neg, float& n) {
  float ax = fabsf(x);
  float lg = __logf(1.0f + __expf(-ax));        // softplus(-|x|)
  float logp   = (x >= 0.0f) ? -lg : (x - lg);  // log sigmoid(x)
  float log1mp = logp - x;                      // log(1 - sigmoid(x))
  float p = __expf(logp);
  float q = __expf(log1mp);                     // (1 - p), accurate
  if (t == 1.0f) {
    pos += q * q * logp;
    n   += 1.0f;
  } else {
    float u = 1.0f - t;
    u = u * u; u = u * u;                       // (1-t)^4
    neg += u * p * p * log1mp;
  }
}

__global__ void zero_ws(float* ws) { ws[threadIdx.x] = 0.0f; }

// Grid: (BLKX, BATCH, 2). ws layout: [s*16+b]*3 -> {pos, neg, n}.
__global__ __launch_bounds__(TPB) void focal_kernel(
    const float* __restrict__ t_tl, const float* __restrict__ t_br,
    const float* __restrict__ p_tl, const float* __restrict__ p_br,
    float* __restrict__ ws) {
  const int b = blockIdx.y;
  const int s = blockIdx.z;
  const float4* T4 = (const float4*)((s == 0) ? t_tl : t_br) + (size_t)b * N4PB;
  const float4* P4 = (const float4*)((s == 0) ? p_tl : p_br) + (size_t)b * N4PB;

  const int tid  = threadIdx.x;
  const int lane = tid & 31;
  const int wave = tid >> 5;
  const int g    = blockIdx.x * TPB + tid;   // 0..STRIDE4-1

  __shared__ float4 sT[TPB / 32][2][32];
  __shared__ float4 sP[TPB / 32][2][32];
  uint32_t ldsT[2], ldsP[2];
  ldsT[0] = (uint32_t)(uintptr_t)&sT[wave][0][lane];
  ldsT[1] = (uint32_t)(uintptr_t)&sT[wave][1][lane];
  ldsP[0] = (uint32_t)(uintptr_t)&sP[wave][0][lane];
  ldsP[1] = (uint32_t)(uintptr_t)&sP[wave][1][lane];

  auto issue = [&](int it, int slot) {
    const float4* gt = T4 + (size_t)it * STRIDE4 + g;
    const float4* gp = P4 + (size_t)it * STRIDE4 + g;
    async_b128(ldsT[slot], (const void*)gt);
    async_b128(ldsP[slot], (const void*)gp);
  };

  float pos = 0.0f, neg = 0.0f, n = 0.0f;
  issue(0, 0);
  for (int it = 0; it < ITERS; ++it) {
    const int slot = it & 1;
    if (it + 1 < ITERS) {
      issue(it + 1, slot ^ 1);
      wait_async_le2();
    } else {
      wait_async_0();
    }
    float4 t4 = sT[wave][slot][lane];
    float4 p4 = sP[wave][slot][lane];
    focal_elem(t4.x, p4.x, pos, neg, n);
    focal_elem(t4.y, p4.y, pos, neg, n);
    focal_elem(t4.z, p4.z, pos, neg, n);
    focal_elem(t4.w, p4.w, pos, neg, n);
  }

  pos = wave_sum32(pos);
  neg = wave_sum32(neg);
  n   = wave_sum32(n);

  __shared__ float red[3][TPB / 32];
  if (lane == 0) { red[0][wave] = pos; red[1][wave] = neg; red[2][wave] = n; }
  __syncthreads();
  if (tid == 0) {
    float P0 = 0.0f, N0 = 0.0f, C0 = 0.0f;
    for (int w = 0; w < TPB / 32; ++w) {
      P0 += red[0][w]; N0 += red[1][w]; C0 += red[2][w];
    }
    float* acc = ws + (size_t)(s * BATCH + b) * 3;
    atomicAdd(acc + 0, P0);
    atomicAdd(acc + 1, N0);
    atomicAdd(acc + 2, C0);
  }
}

__device__ __forceinline__ float smooth_l1(float d) {
  float ad = fabsf(d);
  return (ad < 1.0f) ? (0.5f * d * d) : (ad - 0.5f);
}

// ws[96]=sum(mask), ws[97]=l_tl, ws[98]=l_br, ws[99]=pull, ws[100]=push
__global__ __launch_bounds__(KK) void pair_kernel(
    const float* __restrict__ t_tl_off, const float* __restrict__ t_br_off,
    const float* __restrict__ p_tl_off, const float* __restrict__ p_br_off,
    const float* __restrict__ p_tl_emb, const float* __restrict__ p_br_emb,
    const int* __restrict__ idx_tl, const int* __restrict__ idx_br,
    const int* __restrict__ mask, float* __restrict__ ws) {
  const int b = blockIdx.x;
  const int k = threadIdx.x;

  const int itl = idx_tl[b * KK + k];
  const int ibr = idx_br[b * KK + k];
  const float m = (float)mask[b * KK + k];

  const size_t ob = (size_t)b * 2 * HWSZ;
  float pt0 = p_tl_off[ob + 0 * HWSZ + itl];
  float pt1 = p_tl_off[ob + 1 * HWSZ + itl];
  float pb0 = p_br_off[ob + 0 * HWSZ + ibr];
  float pb1 = p_br_off[ob + 1 * HWSZ + ibr];
  float tt0 = t_tl_off[((size_t)b * KK + k) * 2 + 0];
  float tt1 = t_tl_off[((size_t)b * KK + k) * 2 + 1];
  float tb0 = t_br_off[((size_t)b * KK + k) * 2 + 0];
  float tb1 = t_br_off[((size_t)b * KK + k) * 2 + 1];
  float l_tl = (smooth_l1(pt0 - tt0) + smooth_l1(pt1 - tt1)) * m;
  float l_br = (smooth_l1(pb0 - tb0) + smooth_l1(pb1 - tb1)) * m;

  float tle = p_tl_emb[(size_t)b * HWSZ + itl];
  float bre = p_br_emb[(size_t)b * HWSZ + ibr];
  float ek  = 0.5f * (tle + bre);

  __shared__ float s_m[KK], s_ek[KK];
  s_m[k] = m; s_ek[k] = ek;
  __syncthreads();

  float num = 0.0f, relu_sum = 0.0f;
  for (int j = 0; j < KK; ++j) {
    float mj = s_m[j];
    num += mj;
    if (m == 1.0f && mj == 1.0f) {
      float r = 2.0f - fabsf(s_ek[j] - ek);
      relu_sum += (r > 0.0f) ? r : 0.0f;
    }
  }
  float inv  = 1.0f / (num + EPSV);
  float pull = ((tle - ek) * (tle - ek) + (bre - ek) * (bre - ek)) * inv * m;

  float r0 = wave_sum32(l_tl);
  float r1 = wave_sum32(l_br);
  float r2 = wave_sum32(pull);
  float r3 = wave_sum32(relu_sum);

  __shared__ float red[4][KK / 32];
  const int lane = k & 31, wave = k >> 5;
  if (lane == 0) { red[0][wave] = r0; red[1][wave] = r1; red[2][wave] = r2; red[3][wave] = r3; }
  __syncthreads();
  if (k == 0) {
    float a0 = 0, a1 = 0, a2 = 0, a3 = 0;
    for (int w = 0; w < KK / 32; ++w) {
      a0 += red[0][w]; a1 += red[1][w]; a2 += red[2][w]; a3 += red[3][w];
    }
    // push = sum_pairs[relu - 2/(num+eps)] / ((num-1)*num + eps);
    // pair count = num*num (diagonal of masked entries included).
    float num2 = (num - 1.0f) * num;
    float push = (a3 - num * num * (2.0f * inv)) / (num2 + EPSV);
    atomicAdd(ws + 96,  num);
    atomicAdd(ws + 97,  a0);
    atomicAdd(ws + 98,  a1);
    atomicAdd(ws + 99,  a2);
    atomicAdd(ws + 100, push);
  }
}

__global__ void final_kernel(const float* __restrict__ ws,
                             float* __restrict__ out) {
  const int t = threadIdx.x;  // 32 threads, one wave
  float per_tl = 0.0f, per_br = 0.0f;
  if (t < BATCH) {
    const float* a = ws + t * 3;
    float p = a[0], ng = a[1], n = a[2];
    per_tl = (n == 0.0f) ? ng : (p + ng) / n;
    const float* c = ws + 3 * BATCH + t * 3;
    p = c[0]; ng = c[1]; n = c[2];
    per_br = (n == 0.0f) ? ng : (p + ng) / n;
  }
  float stl = wave_sum32(per_tl);
  float sbr = wave_sum32(per_br);
  if (t == 0) {
    float det  = 0.5f * (-stl - sbr);
    float numx = ws[96] * 2.0f;
    float off  = ws[97] / (numx + EPSV) + ws[98] / (numx + EPSV);
    out[0] = (det + ws[99] + ws[100] + off) / (float)BATCH;
  }
}

extern "C" void kernel_launch(void* const* d_in, const int* in_sizes, int n_in,
                              void* d_out, int out_size, void* d_ws,
                              size_t ws_size, hipStream_t stream) {
  const float* t_tl_heat = (const float*)d_in[0];
  const float* t_br_heat = (const float*)d_in[1];
  const float* t_tl_off  = (const float*)d_in[2];
  const float* t_br_off  = (const float*)d_in[3];
  const float* p_tl_heat = (const float*)d_in[4];
  const float* p_br_heat = (const float*)d_in[5];
  const float* p_tl_off  = (const float*)d_in[6];
  const float* p_br_off  = (const float*)d_in[7];
  const float* p_tl_emb  = (const float*)d_in[8];
  const float* p_br_emb  = (const float*)d_in[9];
  const int*   idx_tl    = (const int*)d_in[10];
  const int*   idx_br    = (const int*)d_in[11];
  const int*   mask      = (const int*)d_in[12];
  float* ws  = (float*)d_ws;
  float* out = (float*)d_out;

  zero_ws<<<1, 128, 0, stream>>>(ws);
  dim3 grid(BLKX, BATCH, 2);
  focal_kernel<<<grid, TPB, 0, stream>>>(t_tl_heat, t_br_heat, p_tl_heat,
                                         p_br_heat, ws);
  pair_kernel<<<BATCH, KK, 0, stream>>>(t_tl_off, t_br_off, p_tl_off, p_br_off,
                                        p_tl_emb, p_br_emb, idx_tl, idx_br,
                                        mask, ws);
  final_kernel<<<1, 32, 0, stream>>>(ws, out);
}